// featureGenerationNet2_49435073577543
// MI455X (gfx1250) — compile-verified
//
#include <hip/hip_runtime.h>
#include <hip/hip_bf16.h>
#include <cmath>

// ---------------------------------------------------------------------------
// featureGenerationNet2: GraphConv x4 (1->20->15->12->10) + FC x4 (10->32->16->8->16)
// Strategy (MI455X):
//   * edge scatter: edge-per-thread, native f32 L2 atomics (unsafeAtomicAdd ->
//     global_atomic_add_f32). Feature matrices (<=25.6MB) are L2-resident
//     (192MB L2), so gathers+atomics never touch HBM after first pass.
//   * node linears: V_WMMA_F32_16X16X4_F32 (exact fp32 matrix path).
//     One wave == one 16-node tile. deg_inv scaling fused into A-load,
//     bias+softplus fused into D-store.
// ---------------------------------------------------------------------------

typedef __attribute__((ext_vector_type(2))) float v2f;
typedef __attribute__((ext_vector_type(8))) float v8f;

__device__ __forceinline__ float softplus_f(float v) {
    // numerically stable softplus matching jax.nn.softplus
    return (v > 20.0f) ? v : log1pf(expf(v));
}

// ---------------------------------------------------------------------------
__global__ void zero_kernel(float* __restrict__ p, int n) {
    int i = blockIdx.x * blockDim.x + threadIdx.x;
    if (i < n) p[i] = 0.0f;
}

__global__ void degree_kernel(const long long* __restrict__ tgt,
                              float* __restrict__ cnt, int E) {
    int e = blockIdx.x * blockDim.x + threadIdx.x;
    if (e < E) unsafeAtomicAdd(&cnt[(int)tgt[e]], 1.0f);
}

__global__ void deginv_kernel(const float* __restrict__ cnt,
                              float* __restrict__ deg_inv, int n) {
    int i = blockIdx.x * blockDim.x + threadIdx.x;
    if (i < n) deg_inv[i] = 1.0f / fmaxf(cnt[i], 1.0f);
}

// scatter-add: agg[tgt] += x[src], F features per edge
template <int F>
__global__ void scatter_add_kernel(const long long* __restrict__ src,
                                   const long long* __restrict__ tgt,
                                   const float* __restrict__ x,
                                   float* __restrict__ agg, int E) {
    int e = blockIdx.x * blockDim.x + threadIdx.x;
    if (e >= E) return;
    int s = (int)src[e];
    int t = (int)tgt[e];
#pragma unroll
    for (int f = 0; f < F; ++f)
        unsafeAtomicAdd(&agg[t * F + f], x[s * F + f]);
}

// ---------------------------------------------------------------------------
// Node-level linear via fp32 WMMA.
//   GCN:  out = softplus( (A*deg_inv) @ W1^T + bias + X @ W2^T )
//   FC :  out = [softplus]( A @ W1^T + bias )
// A,X: [n,FI] row-major; W1,W2: [FO,FI] row-major; out: [n,FO].
// One wave handles 16 nodes. K padded to mult of 4, N-tiles of 16 columns.
//
// ISA layouts (CDNA5 ISA §7.12.2, 32-bit, wave32):
//   A 16x4 : lane l holds A[l&15][2*(l>>4)+j] in vgpr j
//   B 4x16 : lane l holds B[2*(l>>4)+j][l&15] in vgpr j
//   C/D    : vgpr v, lane l -> D[v + 8*(l>>4)][l&15]
// ---------------------------------------------------------------------------
template <int FI, int FO, bool GCN, bool SP>
__global__ void node_linear_wmma_kernel(const float* __restrict__ A,
                                        const float* __restrict__ deg_inv,
                                        const float* __restrict__ X,
                                        const float* __restrict__ W1,
                                        const float* __restrict__ bias,
                                        const float* __restrict__ W2,
                                        float* __restrict__ out, int n_nodes) {
    constexpr int KT = (FI + 3) / 4;    // K chunks of 4
    constexpr int NT = (FO + 15) / 16;  // output column tiles of 16

    const int lane = threadIdx.x & 31;
    const int wave = threadIdx.x >> 5;
    const int tile = blockIdx.x * (blockDim.x >> 5) + wave;
    const int base = tile * 16;
    if (base >= n_nodes) return;  // wave-uniform exit (EXEC all-1s for WMMA)

    const int m    = lane & 15;
    const int half = lane >> 4;
    const int node_a = base + m;
    const bool arow  = node_a < n_nodes;

    float scale = 1.0f;
    if constexpr (GCN) { if (arow) scale = deg_inv[node_a]; }

    // Preload A fragments (zero-padded past FI / past n_nodes)
    float a0[KT][2];
    float a1[KT][2];
#pragma unroll
    for (int kc = 0; kc < KT; ++kc) {
#pragma unroll
        for (int j = 0; j < 2; ++j) {
            const int k = kc * 4 + half * 2 + j;
            const bool ok = arow && (k < FI);
            a0[kc][j] = ok ? A[node_a * FI + k] * scale : 0.0f;
            if constexpr (GCN) a1[kc][j] = ok ? X[node_a * FI + k] : 0.0f;
        }
    }

#pragma unroll
    for (int nt = 0; nt < NT; ++nt) {
        const int col = nt * 16 + m;
        const bool colok = col < FO;
        v8f acc = {0.f, 0.f, 0.f, 0.f, 0.f, 0.f, 0.f, 0.f};
#pragma unroll
        for (int kc = 0; kc < KT; ++kc) {
            v2f av; av[0] = a0[kc][0]; av[1] = a0[kc][1];
            v2f bv;
#pragma unroll
            for (int j = 0; j < 2; ++j) {
                const int k = kc * 4 + half * 2 + j;
                bv[j] = (colok && k < FI) ? W1[col * FI + k] : 0.0f;
            }
            acc = __builtin_amdgcn_wmma_f32_16x16x4_f32(
                false, av, false, bv, (short)0, acc, false, false);
            if constexpr (GCN) {
                v2f av2; av2[0] = a1[kc][0]; av2[1] = a1[kc][1];
                v2f bv2;
#pragma unroll
                for (int j = 0; j < 2; ++j) {
                    const int k = kc * 4 + half * 2 + j;
                    bv2[j] = (colok && k < FI) ? W2[col * FI + k] : 0.0f;
                }
                acc = __builtin_amdgcn_wmma_f32_16x16x4_f32(
                    false, av2, false, bv2, (short)0, acc, false, false);
            }
        }
        if (colok) {
            const float bb = bias[col];
#pragma unroll
            for (int v = 0; v < 8; ++v) {
                const int nd = base + v + half * 8;
                if (nd < n_nodes) {
                    float val = acc[v] + bb;
                    if constexpr (SP) val = softplus_f(val);
                    out[nd * FO + col] = val;
                }
            }
        }
    }
}

// ---------------------------------------------------------------------------
extern "C" void kernel_launch(void* const* d_in, const int* in_sizes, int n_in,
                              void* d_out, int out_size, void* d_ws, size_t ws_size,
                              hipStream_t stream) {
    const int N = in_sizes[0];      // 200,000 (x is [N,1])
    const int E = in_sizes[1] / 2;  // 4,000,000

    const float*     x    = (const float*)d_in[0];
    const long long* ei   = (const long long*)d_in[1];
    const long long* srcI = ei;       // row 0
    const long long* tgtI = ei + E;   // row 1

    const float* g1Wr = (const float*)d_in[2];
    const float* g1b  = (const float*)d_in[3];
    const float* g1Wl = (const float*)d_in[4];
    const float* g2Wr = (const float*)d_in[5];
    const float* g2b  = (const float*)d_in[6];
    const float* g2Wl = (const float*)d_in[7];
    const float* g3Wr = (const float*)d_in[8];
    const float* g3b  = (const float*)d_in[9];
    const float* g3Wl = (const float*)d_in[10];
    const float* g4Wr = (const float*)d_in[11];
    const float* g4b  = (const float*)d_in[12];
    const float* g4Wl = (const float*)d_in[13];
    const float* f1W  = (const float*)d_in[14];
    const float* f1b  = (const float*)d_in[15];
    const float* f2W  = (const float*)d_in[16];
    const float* f2b  = (const float*)d_in[17];
    const float* f3W  = (const float*)d_in[18];
    const float* f3b  = (const float*)d_in[19];
    const float* f4W  = (const float*)d_in[20];
    const float* f4b  = (const float*)d_in[21];

    // workspace layout (floats)
    float* ws      = (float*)d_ws;
    float* cnt     = ws;            // N
    float* deg_inv = ws + (size_t)N;        // N
    float* agg     = ws + (size_t)2  * N;   // N*20 (reused per layer)
    float* h1      = ws + (size_t)22 * N;   // N*20
    float* h2      = ws + (size_t)42 * N;   // N*15
    float* h3      = ws + (size_t)57 * N;   // N*12
    float* h4      = ws + (size_t)69 * N;   // N*10
    float* fA      = ws + (size_t)79 * N;   // N*32
    float* fB      = ws + (size_t)111 * N;  // N*16
    float* fC      = ws + (size_t)127 * N;  // N*8
    float* outp    = (float*)d_out;         // N*16

    const int TB = 256;
    auto gridN = [&](int n) { return (n + TB - 1) / TB; };
    const int gridE = (E + TB - 1) / TB;
    // linear kernels: 8 waves/block -> 128 nodes per block
    const int tiles  = (N + 15) / 16;
    const int gridL  = (tiles + 7) / 8;

    // degrees (recomputed every call: deterministic, no cross-call state)
    zero_kernel<<<gridN(N), TB, 0, stream>>>(cnt, N);
    degree_kernel<<<gridE, TB, 0, stream>>>(tgtI, cnt, E);
    deginv_kernel<<<gridN(N), TB, 0, stream>>>(cnt, deg_inv, N);

    // ---- GCN layer 1: 1 -> 20 ----
    zero_kernel<<<gridN(N * 1), TB, 0, stream>>>(agg, N * 1);
    scatter_add_kernel<1><<<gridE, TB, 0, stream>>>(srcI, tgtI, x, agg, E);
    node_linear_wmma_kernel<1, 20, true, true><<<gridL, TB, 0, stream>>>(
        agg, deg_inv, x, g1Wr, g1b, g1Wl, h1, N);

    // ---- GCN layer 2: 20 -> 15 ----
    zero_kernel<<<gridN(N * 20), TB, 0, stream>>>(agg, N * 20);
    scatter_add_kernel<20><<<gridE, TB, 0, stream>>>(srcI, tgtI, h1, agg, E);
    node_linear_wmma_kernel<20, 15, true, true><<<gridL, TB, 0, stream>>>(
        agg, deg_inv, h1, g2Wr, g2b, g2Wl, h2, N);

    // ---- GCN layer 3: 15 -> 12 ----
    zero_kernel<<<gridN(N * 15), TB, 0, stream>>>(agg, N * 15);
    scatter_add_kernel<15><<<gridE, TB, 0, stream>>>(srcI, tgtI, h2, agg, E);
    node_linear_wmma_kernel<15, 12, true, true><<<gridL, TB, 0, stream>>>(
        agg, deg_inv, h2, g3Wr, g3b, g3Wl, h3, N);

    // ---- GCN layer 4: 12 -> 10 ----
    zero_kernel<<<gridN(N * 12), TB, 0, stream>>>(agg, N * 12);
    scatter_add_kernel<12><<<gridE, TB, 0, stream>>>(srcI, tgtI, h3, agg, E);
    node_linear_wmma_kernel<12, 10, true, true><<<gridL, TB, 0, stream>>>(
        agg, deg_inv, h3, g4Wr, g4b, g4Wl, h4, N);

    // ---- FC tail ----
    node_linear_wmma_kernel<10, 32, false, true><<<gridL, TB, 0, stream>>>(
        h4, nullptr, nullptr, f1W, f1b, nullptr, fA, N);
    node_linear_wmma_kernel<32, 16, false, true><<<gridL, TB, 0, stream>>>(
        fA, nullptr, nullptr, f2W, f2b, nullptr, fB, N);
    node_linear_wmma_kernel<16, 8, false, true><<<gridL, TB, 0, stream>>>(
        fB, nullptr, nullptr, f3W, f3b, nullptr, fC, N);
    node_linear_wmma_kernel<8, 16, false, false><<<gridL, TB, 0, stream>>>(
        fC, nullptr, nullptr, f4W, f4b, nullptr, outp, N);

    (void)n_in; (void)out_size; (void)ws_size;
}